// EncoderDecoderWithAttention_90709709292214
// MI455X (gfx1250) — compile-verified
//
#include <hip/hip_runtime.h>
#include <math.h>
#include <stdint.h>

// Problem constants (from setup_inputs)
#define H       2048
#define H3      6144
#define IN_DIM  128
#define OUT_DIM 128
#define T_ENC   512
#define T_DEC   64

typedef float v2f __attribute__((ext_vector_type(2)));
typedef float v8f __attribute__((ext_vector_type(8)));

#if defined(__has_builtin)
#if __has_builtin(__builtin_amdgcn_global_load_async_to_lds_b128)
#define ASYNC_LDS_OK 1
#endif
#endif

#ifdef ASYNC_LDS_OK
// Exact parameter types per hipcc diagnostic: pointer to 128-bit int vector,
// global AS for the source, LDS AS for the destination.
typedef int b128_vec __attribute__((vector_size(16)));
typedef b128_vec __attribute__((address_space(1)))* g128_ptr;
typedef b128_vec __attribute__((address_space(3)))* l128_ptr;
#endif

__device__ __forceinline__ float wave_sum(float s) {
#pragma unroll
  for (int off = 16; off > 0; off >>= 1) s += __shfl_xor(s, off, 32);
  return s;
}
__device__ __forceinline__ float wave_max(float m) {
#pragma unroll
  for (int off = 16; off > 0; off >>= 1) m = fmaxf(m, __shfl_xor(m, off, 32));
  return m;
}
__device__ __forceinline__ float sigmoidf(float x) { return 1.0f / (1.0f + expf(-x)); }

// ---------------------------------------------------------------------------
// Stage a shared activation vector into LDS once per block.
// Preferred path: gfx1250 async copy engine (GLOBAL_LOAD_ASYNC_TO_LDS_B128,
// tracked by ASYNCcnt) so the VMEM pipe is left free to stream weights.
// Low 32 bits of a generic LDS pointer ARE the LDS byte offset (aperture
// layout), so the int-trunc cast to addrspace(3) is value-correct.
// Must be called by ALL threads of the block (contains a barrier).
// nfloats must be a multiple of 4; pointers 16B-aligned.
// ---------------------------------------------------------------------------
__device__ __forceinline__ void stage_to_lds(float* dst, const float* src, int nfloats) {
  int t = threadIdx.x, nt = blockDim.x;
#ifdef ASYNC_LDS_OK
  for (int i = t * 4; i < nfloats; i += nt * 4)
    __builtin_amdgcn_global_load_async_to_lds_b128(
        (g128_ptr)(uintptr_t)(src + i),
        (l128_ptr)(uint32_t)(uintptr_t)(dst + i),
        0, 0);
#if __has_builtin(__builtin_amdgcn_s_wait_asynccnt)
  __builtin_amdgcn_s_wait_asynccnt(0);
#else
  asm volatile("s_wait_asynccnt 0x0" ::: "memory");
#endif
#else
  for (int i = t * 4; i < nfloats; i += nt * 4)
    *(float4*)(dst + i) = *(const float4*)(src + i);
#endif
  __syncthreads();
}

// ---------------------------------------------------------------------------
// init: zero h0 (row 0 of hbuf) and set prev-argmax index to -1 (=> emb = bias)
// ---------------------------------------------------------------------------
__global__ void init_kernel(float* hbuf, int* idx) {
  int i = blockIdx.x * blockDim.x + threadIdx.x;
  if (i < H) hbuf[i] = 0.0f;
  if (i == 0) *idx = -1;
}

// ---------------------------------------------------------------------------
// WMMA fp32 GEMM:  C[M x N] = A[M x K] @ W[N x K]^T + bias[N]
// One wave per 16x16 C tile, K-loop in steps of 4 using V_WMMA_F32_16X16X4_F32.
// (codegen-verified round 1: unrolls x8 with b64 loads + loadcnt interleave)
// ---------------------------------------------------------------------------
__global__ void wmma_gemm_kernel(const float* __restrict__ A,
                                 const float* __restrict__ W,
                                 const float* __restrict__ bias,
                                 float* __restrict__ C,
                                 int M, int N, int K) {
  int wave = (blockIdx.x * blockDim.x + threadIdx.x) >> 5;
  int lane = threadIdx.x & 31;
  int tiles_n = N >> 4;
  int tm = (wave / tiles_n) << 4;
  int tn = (wave % tiles_n) << 4;
  if (tm >= M) return;

  int half = lane >> 4;   // 0: K pair {0,1}; 1: K pair {2,3}
  int l16  = lane & 15;

  const float* arow = A + (size_t)(tm + l16) * K + 2 * half;
  const float* wrow = W + (size_t)(tn + l16) * K + 2 * half;

  v8f acc = {};
  for (int kb = 0; kb < K; kb += 4) {
    v2f a, b;
    a.x = arow[kb];     a.y = arow[kb + 1];
    b.x = wrow[kb];     b.y = wrow[kb + 1];
    acc = __builtin_amdgcn_wmma_f32_16x16x4_f32(
        false, a, false, b, (short)0, acc, false, false);
  }

  float bv = bias[tn + l16];
#pragma unroll
  for (int i = 0; i < 8; ++i) {
    int row = tm + half * 8 + i;
    C[(size_t)row * N + (tn + l16)] = acc[i] + bv;
  }
}

// ---------------------------------------------------------------------------
// Mat-vec: y[r] = W[r,:] . x + b[r].  Wave per row, 8 waves/block.
// x staged in LDS (async); W streamed with b128 loads (bandwidth-bound).
// Grid must be exactly R/8 blocks; K multiple of 128; dyn-LDS = K floats.
// ---------------------------------------------------------------------------
__global__ void matvec_kernel(const float* __restrict__ W,
                              const float* __restrict__ x,
                              const float* __restrict__ b,
                              float* __restrict__ y, int K) {
  extern __shared__ float xs[];
  stage_to_lds(xs, x, K);
  int wave = (blockIdx.x * blockDim.x + threadIdx.x) >> 5;
  int lane = threadIdx.x & 31;
  const float4* wr = (const float4*)(W + (size_t)wave * K);
  int K4 = K >> 2;
  float s = 0.0f;
  for (int k = lane; k < K4; k += 32) {
    float4 w = wr[k];
    float4 xv = *(const float4*)(xs + 4 * k);
    s += w.x * xv.x + w.y * xv.y + w.z * xv.z + w.w * xv.w;
  }
  s = wave_sum(s);
  if (lane == 0) y[wave] = s + b[wave];
}

// ---------------------------------------------------------------------------
// Fused GRU step. hprev staged in LDS (async). Block owns 8 hidden units
// (24 gate rows); wave w computes the 3 gate rows of unit j0+w with b128
// weight streaming. Launch-per-step = scan synchronization. Grid = H/8.
// ---------------------------------------------------------------------------
__global__ void gru_step_kernel(const float* __restrict__ Whh,
                                const float* __restrict__ bhh,
                                const float* __restrict__ gi,
                                const float* __restrict__ hprev,
                                float* __restrict__ hnext) {
  __shared__ float hsh[H];
  __shared__ float ghs[24];
  stage_to_lds(hsh, hprev, H);

  int tid = threadIdx.x, lane = tid & 31, wid = tid >> 5;
  int j0 = blockIdx.x * 8;
  int j = j0 + wid;
#pragma unroll
  for (int g = 0; g < 3; ++g) {
    const float4* wr = (const float4*)(Whh + (size_t)(g * H + j) * H);
    float s = 0.0f;
    for (int k = lane; k < (H >> 2); k += 32) {
      float4 w = wr[k];
      float4 hv = *(const float4*)(hsh + 4 * k);
      s += w.x * hv.x + w.y * hv.y + w.z * hv.z + w.w * hv.w;
    }
    s = wave_sum(s);
    if (lane == 0) ghs[wid * 3 + g] = s;
  }
  __syncthreads();
  if (tid < 8) {
    int jj = j0 + tid;
    float gh_r = ghs[tid * 3 + 0] + bhh[jj];
    float gh_z = ghs[tid * 3 + 1] + bhh[H + jj];
    float gh_n = ghs[tid * 3 + 2] + bhh[2 * H + jj];
    float rg = sigmoidf(gi[jj] + gh_r);
    float zg = sigmoidf(gi[H + jj] + gh_z);
    float ng = tanhf(gi[2 * H + jj] + rg * gh_n);
    hnext[jj] = (1.0f - zg) * ng + zg * hsh[jj];
  }
}

// ---------------------------------------------------------------------------
// scores[s] = tanh(U[s,:] + wd) . attn_W + attn_b   (wave per row; grid 64x256)
// wd and attn_W staged in LDS; U streamed b128.
// ---------------------------------------------------------------------------
__global__ void attn_scores_kernel(const float* __restrict__ U,
                                   const float* __restrict__ wd,
                                   const float* __restrict__ attnW,
                                   const float* __restrict__ attnB,
                                   float* __restrict__ scores) {
  __shared__ float wds[H];
  __shared__ float aws[H];
  stage_to_lds(wds, wd, H);
  stage_to_lds(aws, attnW, H);

  int wave = (blockIdx.x * blockDim.x + threadIdx.x) >> 5;
  int lane = threadIdx.x & 31;
  const float4* ur = (const float4*)(U + (size_t)wave * H);
  float s = 0.0f;
  for (int k = lane; k < (H >> 2); k += 32) {
    float4 u = ur[k];
    float4 w = *(const float4*)(wds + 4 * k);
    float4 a = *(const float4*)(aws + 4 * k);
    s += tanhf(u.x + w.x) * a.x + tanhf(u.y + w.y) * a.y +
         tanhf(u.z + w.z) * a.z + tanhf(u.w + w.w) * a.w;
  }
  s = wave_sum(s);
  if (lane == 0) scores[wave] = s + attnB[0];
}

// ---------------------------------------------------------------------------
// softmax over 512 scores (one 512-thread block, 16 waves)
// ---------------------------------------------------------------------------
__global__ void softmax512_kernel(const float* __restrict__ scores,
                                  float* __restrict__ aw,
                                  float* __restrict__ attn_out) {
  __shared__ float redm[16];
  __shared__ float reds[16];
  __shared__ float bc[2];
  int tid = threadIdx.x, lane = tid & 31, wid = tid >> 5;
  float v = scores[tid];
  float m = wave_max(v);
  if (lane == 0) redm[wid] = m;
  __syncthreads();
  if (tid == 0) {
    float mm = redm[0];
    for (int w = 1; w < 16; ++w) mm = fmaxf(mm, redm[w]);
    bc[0] = mm;
  }
  __syncthreads();
  float M = bc[0];
  float e = expf(v - M);
  float s = wave_sum(e);
  if (lane == 0) reds[wid] = s;
  __syncthreads();
  if (tid == 0) {
    float ss = 0.0f;
    for (int w = 0; w < 16; ++w) ss += reds[w];
    bc[1] = ss;
  }
  __syncthreads();
  float a = e / bc[1];
  aw[tid] = a;
  attn_out[tid] = a;
}

// ---------------------------------------------------------------------------
// ctx[j] = sum_s aw[s] * enc_out[s][j]; aw staged in LDS (broadcast reads),
// float4 per thread over j. Grid = H/1024 blocks x 256.
// ---------------------------------------------------------------------------
__global__ void ctx_kernel(const float* __restrict__ aw,
                           const float* __restrict__ hbuf,
                           float* __restrict__ ctx) {
  __shared__ float aws[T_ENC];
  stage_to_lds(aws, aw, T_ENC);
  int j4 = (blockIdx.x * blockDim.x + threadIdx.x) * 4;
  float4 acc = {0.0f, 0.0f, 0.0f, 0.0f};
  for (int t = 0; t < T_ENC; ++t) {
    float a = aws[t];
    float4 hv = *(const float4*)(hbuf + (size_t)(t + 1) * H + j4);
    acc.x += a * hv.x; acc.y += a * hv.y; acc.z += a * hv.z; acc.w += a * hv.w;
  }
  *(float4*)(ctx + j4) = acc;
}

// ---------------------------------------------------------------------------
// din = [emb, ctx]; inp is one-hot(argmax) => emb[j] = o2h_W[j][idx] + o2h_b[j]
// ---------------------------------------------------------------------------
__global__ void din_kernel(const float* __restrict__ o2hW,
                           const float* __restrict__ o2hb,
                           const float* __restrict__ ctx,
                           const int* __restrict__ idx,
                           float* __restrict__ din) {
  int j = blockIdx.x * blockDim.x + threadIdx.x;
  if (j >= H) return;
  int id = *idx;
  float emb = o2hb[j];
  if (id >= 0) emb += o2hW[(size_t)j * OUT_DIM + id];
  din[j] = emb;
  din[H + j] = ctx[j];
}

// ---------------------------------------------------------------------------
// log_softmax(128) + argmax (first-index tie-break, matches jnp.argmax)
// ---------------------------------------------------------------------------
__global__ void logsoftmax_argmax_kernel(const float* __restrict__ logits,
                                         float* __restrict__ out_row,
                                         int* __restrict__ idx) {
  __shared__ float rm[4], rs[4], rv[4];
  __shared__ int ri[4];
  int tid = threadIdx.x, lane = tid & 31, wid = tid >> 5;
  float v = logits[tid];
  float m = wave_max(v);
  if (lane == 0) rm[wid] = m;
  __syncthreads();
  float M = fmaxf(fmaxf(rm[0], rm[1]), fmaxf(rm[2], rm[3]));
  float e = expf(v - M);
  float s = wave_sum(e);
  if (lane == 0) rs[wid] = s;
  __syncthreads();
  float S = rs[0] + rs[1] + rs[2] + rs[3];
  out_row[tid] = v - M - logf(S);
  float bv = v;
  int bi = tid;
#pragma unroll
  for (int off = 16; off > 0; off >>= 1) {
    float ov = __shfl_xor(bv, off, 32);
    int oi = __shfl_xor(bi, off, 32);
    if (ov > bv || (ov == bv && oi < bi)) { bv = ov; bi = oi; }
  }
  if (lane == 0) { rv[wid] = bv; ri[wid] = bi; }
  __syncthreads();
  if (tid == 0) {
    float best = rv[0];
    int besti = ri[0];
    for (int w = 1; w < 4; ++w)
      if (rv[w] > best || (rv[w] == best && ri[w] < besti)) { best = rv[w]; besti = ri[w]; }
    *idx = besti;
  }
}

// ---------------------------------------------------------------------------
extern "C" void kernel_launch(void* const* d_in, const int* in_sizes, int n_in,
                              void* d_out, int out_size, void* d_ws, size_t ws_size,
                              hipStream_t stream) {
  const float* x     = (const float*)d_in[0];
  const float* eWih  = (const float*)d_in[1];
  const float* eWhh  = (const float*)d_in[2];
  const float* ebih  = (const float*)d_in[3];
  const float* ebhh  = (const float*)d_in[4];
  const float* dWih  = (const float*)d_in[5];
  const float* dWhh  = (const float*)d_in[6];
  const float* dbih  = (const float*)d_in[7];
  const float* dbhh  = (const float*)d_in[8];
  const float* h2oW  = (const float*)d_in[9];
  const float* h2ob  = (const float*)d_in[10];
  const float* UW    = (const float*)d_in[11];
  const float* Ub    = (const float*)d_in[12];
  const float* WW    = (const float*)d_in[13];
  const float* Wb    = (const float*)d_in[14];
  const float* attnW = (const float*)d_in[15];
  const float* attnB = (const float*)d_in[16];
  const float* o2hW  = (const float*)d_in[17];
  const float* o2hb  = (const float*)d_in[18];

  float* outs  = (float*)d_out;                     // 64 x 128
  float* attns = (float*)d_out + T_DEC * OUT_DIM;   // 64 x 512

  // workspace carve-out (~21 MB of fp32)
  float* ws     = (float*)d_ws;
  float* hbuf   = ws; ws += (size_t)(T_ENC + 1) * H;  // h history == enc_out (row t+1)
  float* gi_enc = ws; ws += (size_t)T_ENC * H3;
  float* Umat   = ws; ws += (size_t)T_ENC * H;
  float* wd     = ws; ws += H;
  float* scores = ws; ws += T_ENC;
  float* aw     = ws; ws += T_ENC;
  float* ctx    = ws; ws += H;
  float* din    = ws; ws += 2 * H;
  float* gi_dec = ws; ws += H3;
  float* logits = ws; ws += OUT_DIM;
  float* hA     = ws; ws += H;
  float* hB     = ws; ws += H;
  int*   idx    = (int*)ws;

  init_kernel<<<8, 256, 0, stream>>>(hbuf, idx);

  // GEMM 1 (WMMA): gi_enc = x @ enc_Wih^T + bih   (512 x 6144 x 128)
  wmma_gemm_kernel<<<(T_ENC / 16) * (H3 / 16) / 8, 256, 0, stream>>>(
      x, eWih, ebih, gi_enc, T_ENC, H3, IN_DIM);

  // encoder scan: 512 sequential fused GRU steps (kernel boundary = scan sync)
  for (int t = 0; t < T_ENC; ++t)
    gru_step_kernel<<<H / 8, 256, 0, stream>>>(
        eWhh, ebhh, gi_enc + (size_t)t * H3,
        hbuf + (size_t)t * H, hbuf + (size_t)(t + 1) * H);

  // GEMM 2 (WMMA): U = enc_out @ U_W^T + U_b     (512 x 2048 x 2048)
  wmma_gemm_kernel<<<(T_ENC / 16) * (H / 16) / 8, 256, 0, stream>>>(
      hbuf + H, UW, Ub, Umat, T_ENC, H, H);

  const float* encLast = hbuf + (size_t)T_ENC * H;
  for (int t = 0; t < T_DEC; ++t) {
    const float* hprev = (t == 0) ? encLast : ((t & 1) ? hA : hB);
    float* hnext = (t & 1) ? hB : hA;

    matvec_kernel<<<H / 8, 256, H * sizeof(float), stream>>>(WW, hprev, Wb, wd, H);
    attn_scores_kernel<<<T_ENC / 8, 256, 0, stream>>>(Umat, wd, attnW, attnB, scores);
    softmax512_kernel<<<1, 512, 0, stream>>>(scores, aw, attns + (size_t)t * T_ENC);
    ctx_kernel<<<H / 1024, 256, 0, stream>>>(aw, hbuf, ctx);
    din_kernel<<<H / 256, 256, 0, stream>>>(o2hW, o2hb, ctx, idx, din);
    matvec_kernel<<<H3 / 8, 256, 2 * H * sizeof(float), stream>>>(dWih, din, dbih, gi_dec, 2 * H);
    gru_step_kernel<<<H / 8, 256, 0, stream>>>(dWhh, dbhh, gi_dec, hprev, hnext);
    matvec_kernel<<<OUT_DIM / 8, 256, H * sizeof(float), stream>>>(h2oW, hnext, h2ob, logits, H);
    logsoftmax_argmax_kernel<<<1, 128, 0, stream>>>(logits, outs + (size_t)t * OUT_DIM, idx);
  }
}